// MPL_17403207483852
// MI455X (gfx1250) — compile-verified
//
#include <hip/hip_runtime.h>

typedef __attribute__((ext_vector_type(2))) float v2f;
typedef __attribute__((ext_vector_type(4))) float v4f;
typedef __attribute__((ext_vector_type(8))) float v8f;

#define NN 10000
#define NE 320000
#define DD 256
#define KK 512   // 2*D
#define NKSTEP 128   // KK/4 k-steps of the 16x16x4 WMMA

// ---------------- zero the scatter accumulator (10.24 MB in d_ws) ------------
__global__ void MPL_zero_kernel(float* __restrict__ p, int n4) {
  int i = blockIdx.x * blockDim.x + threadIdx.x;
  v4f z = {0.f, 0.f, 0.f, 0.f};
  for (; i < n4; i += gridDim.x * blockDim.x)
    ((v4f*)p)[i] = z;
}

// ---------------- repack W [256,512] into B-fragment order -------------------
// Wf[((jt*128 + ks)*32 + lane)] = v2f{ W[jt*16 + (lane&15), 4*ks + 2*(lane>>4)],
//                                      W[jt*16 + (lane&15), 4*ks + 2*(lane>>4) + 1] }
// so the GEMM's B loads become fully coalesced 256B global_load_b64 per k-step.
__global__ void MPL_pack_w(const float* __restrict__ W, float* __restrict__ Wf) {
  const int t = blockIdx.x * blockDim.x + threadIdx.x;   // 0 .. 16*128*32-1
  if (t >= 16 * NKSTEP * 32) return;
  const int lane = t & 31;
  const int ks   = (t >> 5) & (NKSTEP - 1);
  const int jt   = t >> 12;                              // 0..15
  const int n    = lane & 15;
  const int kh   = (lane >> 4) * 2;
  const int row  = jt * 16 + n;
  const int col  = ks * 4 + kh;
  v2f p = *(const v2f*)(W + (size_t)row * KK + col);
  *(v2f*)(Wf + (size_t)t * 2) = p;
}

// ---------------- edge stage: msg = node[src]*edge; atomic scatter by dst ----
// One wave32 per edge. D=256 floats = 64 float4; each lane does 2 float4.
// edge_feats is streamed once -> non-temporal loads (keep L2 for node/reduced).
__global__ void MPL_edge_kernel(const float* __restrict__ node_feats,
                                const float* __restrict__ edge_feats,
                                const int* __restrict__ src,
                                const int* __restrict__ dst,
                                float* __restrict__ reduced) {
  const int lane  = threadIdx.x & 31;
  const int wave  = (blockIdx.x * blockDim.x + threadIdx.x) >> 5;
  const int nwave = (gridDim.x * blockDim.x) >> 5;
  for (int e = wave; e < NE; e += nwave) {
    const int s = src[e];
    const int d = dst[e];
    const v4f* nf = (const v4f*)(node_feats + (size_t)s * DD);
    const v4f* ef = (const v4f*)(edge_feats + (size_t)e * DD);
    float* red = reduced + (size_t)d * DD;
#pragma unroll
    for (int c = 0; c < 2; ++c) {
      const int idx = lane + 32 * c;            // float4 index 0..63
      v4f a = nf[idx];                          // L2-hot gather
      v4f b = __builtin_nontemporal_load(ef + idx);  // streamed once
      v4f m = a * b;
#pragma unroll
      for (int j = 0; j < 4; ++j) {
        // no-return f32 atomic add -> global_atomic_add_f32 (L2-side)
        (void)__hip_atomic_fetch_add(&red[idx * 4 + j], m[j],
                                     __ATOMIC_RELAXED, __HIP_MEMORY_SCOPE_AGENT);
      }
    }
  }
}

// ---------------- GEMM: out = relu([node|reduced] @ W^T + b) -----------------
// V_WMMA_F32_16X16X4_F32. Wave computes a 16x64 strip (4 accumulators).
// K loop split in two halves so the A pointer is loop-invariant (no cndmask),
// B fragments come pre-packed (coalesced b64 loads).
__global__ void __launch_bounds__(256)
MPL_gemm_kernel(const float* __restrict__ node_feats,
                const float* __restrict__ reduced,
                const float* __restrict__ Wf,    // packed B fragments
                const float* __restrict__ bias,
                float* __restrict__ out) {
  const int lane = threadIdx.x & 31;
  const int wv   = threadIdx.x >> 5;
  const int gw   = blockIdx.x * 8 + wv;          // wave-uniform
  if (gw >= 2500) return;                        // uniform branch: EXEC all-1

  const int mt = gw >> 2;                        // 0..624
  const int jg = gw & 3;                         // 0..3
  const int ib = mt * 16;
  const int jb = jg * 64;

  const int nl    = lane & 15;
  const int hi    = lane >> 4;
  const int khalf = hi * 2;                      // K sub-offset {0,2}

  // A fragment sources (row = ib + nl of h = [node_feats | reduced])
  const float* nfr = node_feats + (size_t)(ib + nl) * DD + khalf;
  const float* rdr = reduced    + (size_t)(ib + nl) * DD + khalf;

  // Packed B fragment bases for the 4 j-tiles of this j-group
  const v2f* wf = (const v2f*)Wf;
  const size_t b0b = ((size_t)(jg * 4 + 0) * NKSTEP) * 32 + lane;
  const size_t b1b = ((size_t)(jg * 4 + 1) * NKSTEP) * 32 + lane;
  const size_t b2b = ((size_t)(jg * 4 + 2) * NKSTEP) * 32 + lane;
  const size_t b3b = ((size_t)(jg * 4 + 3) * NKSTEP) * 32 + lane;

  v8f acc0 = {}, acc1 = {}, acc2 = {}, acc3 = {};

  // First half of K: h columns 0..255 come from node_feats
  for (int ks = 0; ks < 64; ++ks) {
    v2f a  = *(const v2f*)(nfr + ks * 4);
    v2f b0 = wf[b0b + (size_t)ks * 32];
    v2f b1 = wf[b1b + (size_t)ks * 32];
    v2f b2 = wf[b2b + (size_t)ks * 32];
    v2f b3 = wf[b3b + (size_t)ks * 32];
    acc0 = __builtin_amdgcn_wmma_f32_16x16x4_f32(false, a, false, b0, (short)0, acc0, false, false);
    acc1 = __builtin_amdgcn_wmma_f32_16x16x4_f32(false, a, false, b1, (short)0, acc1, false, false);
    acc2 = __builtin_amdgcn_wmma_f32_16x16x4_f32(false, a, false, b2, (short)0, acc2, false, false);
    acc3 = __builtin_amdgcn_wmma_f32_16x16x4_f32(false, a, false, b3, (short)0, acc3, false, false);
  }
  // Second half of K: h columns 256..511 come from reduced
  for (int ks = 64; ks < NKSTEP; ++ks) {
    v2f a  = *(const v2f*)(rdr + (ks - 64) * 4);
    v2f b0 = wf[b0b + (size_t)ks * 32];
    v2f b1 = wf[b1b + (size_t)ks * 32];
    v2f b2 = wf[b2b + (size_t)ks * 32];
    v2f b3 = wf[b3b + (size_t)ks * 32];
    acc0 = __builtin_amdgcn_wmma_f32_16x16x4_f32(false, a, false, b0, (short)0, acc0, false, false);
    acc1 = __builtin_amdgcn_wmma_f32_16x16x4_f32(false, a, false, b1, (short)0, acc1, false, false);
    acc2 = __builtin_amdgcn_wmma_f32_16x16x4_f32(false, a, false, b2, (short)0, acc2, false, false);
    acc3 = __builtin_amdgcn_wmma_f32_16x16x4_f32(false, a, false, b3, (short)0, acc3, false, false);
  }

  // Epilogue: bias + relu, non-temporal stores (output not re-read on device)
  const float bb0 = bias[jb + nl];
  const float bb1 = bias[jb + 16 + nl];
  const float bb2 = bias[jb + 32 + nl];
  const float bb3 = bias[jb + 48 + nl];
#pragma unroll
  for (int v = 0; v < 8; ++v) {
    const int m = v + 8 * hi;                    // C/D layout: M = vgpr + 8*hi
    float* orow = out + (size_t)(ib + m) * DD + jb;
    __builtin_nontemporal_store(fmaxf(acc0[v] + bb0, 0.f), orow + nl);
    __builtin_nontemporal_store(fmaxf(acc1[v] + bb1, 0.f), orow + 16 + nl);
    __builtin_nontemporal_store(fmaxf(acc2[v] + bb2, 0.f), orow + 32 + nl);
    __builtin_nontemporal_store(fmaxf(acc3[v] + bb3, 0.f), orow + 48 + nl);
  }
}

extern "C" void kernel_launch(void* const* d_in, const int* in_sizes, int n_in,
                              void* d_out, int out_size, void* d_ws, size_t ws_size,
                              hipStream_t stream) {
  const float* node_feats = (const float*)d_in[0];
  const float* edge_feats = (const float*)d_in[1];
  const int*   src        = (const int*)d_in[2];
  const int*   dst        = (const int*)d_in[3];
  const float* W          = (const float*)d_in[4];
  const float* b          = (const float*)d_in[5];
  float* out     = (float*)d_out;
  float* reduced = (float*)d_ws;                       // N*D floats = 10.24 MB
  float* Wf      = (float*)d_ws + (size_t)NN * DD;     // 0.5 MB packed W

  // 1) reduced = 0
  MPL_zero_kernel<<<2500, 256, 0, stream>>>(reduced, (NN * DD) / 4);
  // 2) repack W into WMMA B-fragment order (one-time 1 MB of traffic)
  MPL_pack_w<<<(16 * NKSTEP * 32) / 256, 256, 0, stream>>>(W, Wf);
  // 3) gather * edge, scatter-sum (one wave per edge: 320000/8 blocks)
  MPL_edge_kernel<<<NE / 8, 256, 0, stream>>>(node_feats, edge_feats, src, dst, reduced);
  // 4) fused concat-GEMM + bias + relu on fp32 WMMA
  MPL_gemm_kernel<<<(2500 + 7) / 8, 256, 0, stream>>>(node_feats, reduced, Wf, b, out);
}